// Involution_CUDA_6451040878738
// MI455X (gfx1250) — compile-verified
//
#include <hip/hip_runtime.h>

typedef __attribute__((ext_vector_type(2))) float v2f;
typedef __attribute__((ext_vector_type(8))) float v8f;

#define CCH   256     // channels
#define CK    64      // C / RED
#define HW    56
#define NPIX  3136    // 56*56
#define BATCH 4
#define KW    7
#define TAPS  49
#define NTILE 196     // NPIX / 16

// ---------------- Kernel 1: conv1 (1x1, C -> C/4), pre-BN --------------------
// hpre[b,k,p] = b1[k] + sum_c w1[k,c] * x[b,c,p]
__global__ void involution_conv1(const float* __restrict__ x,
                                 const float* __restrict__ w1,
                                 const float* __restrict__ b1,
                                 float* __restrict__ hpre) {
    const int idx = blockIdx.x * 256 + threadIdx.x;       // < 4*64*3136 = 802816
    const int b   = idx / (CK * NPIX);
    const int rem = idx - b * (CK * NPIX);
    const int k   = rem / NPIX;
    const int p   = rem - k * NPIX;
    const float* wr = w1 + k * CCH;
    const float* xr = x + (size_t)b * CCH * NPIX + p;
    float acc = b1[k];
#pragma unroll 8
    for (int c = 0; c < CCH; ++c)
        acc += wr[c] * xr[(size_t)c * NPIX];
    hpre[idx] = acc;
}

// ---------------- Kernel 2: deterministic per-channel batch stats ------------
__global__ void involution_bnreduce(const float* __restrict__ hpre,
                                    float* __restrict__ stats) { // [0..63]=sum [64..127]=sumsq
    __shared__ float sbuf[256];
    __shared__ float sbuf2[256];
    const int k = blockIdx.x;                              // 0..63
    float s = 0.f, s2 = 0.f;
    for (int i = threadIdx.x; i < BATCH * NPIX; i += 256) {
        const int b = i / NPIX;
        const int p = i - b * NPIX;
        const float v = hpre[(size_t)(b * CK + k) * NPIX + p];
        s += v; s2 += v * v;
    }
    sbuf[threadIdx.x] = s; sbuf2[threadIdx.x] = s2;
    __syncthreads();
    for (int off = 128; off >= 1; off >>= 1) {
        if (threadIdx.x < (unsigned)off) {
            sbuf[threadIdx.x]  += sbuf[threadIdx.x + off];
            sbuf2[threadIdx.x] += sbuf2[threadIdx.x + off];
        }
        __syncthreads();
    }
    if (threadIdx.x == 0) { stats[k] = sbuf[0]; stats[CK + k] = sbuf2[0]; }
}

// ---------------- Kernel 3: fold BN into scale/shift -------------------------
__global__ void involution_bnfinalize(const float* __restrict__ stats,
                                      const float* __restrict__ gamma,
                                      const float* __restrict__ beta,
                                      float* __restrict__ sclsft) {
    const int k = threadIdx.x;
    if (k < CK) {
        const float invN = 1.0f / (float)(BATCH * NPIX);
        const float mu   = stats[k] * invN;
        const float var  = stats[CK + k] * invN - mu * mu;
        const float sc   = gamma[k] * rsqrtf(var + 1e-5f);
        sclsft[k]      = sc;
        sclsft[CK + k] = beta[k] - mu * sc;
    }
}

// ---------------- helper: 8-tap partial contraction --------------------------
__device__ __forceinline__ float tap_accum(v8f acc, int rbase, int half,
                                           int py, int px,
                                           const float* __restrict__ xpl,
                                           const float* __restrict__ b2c) {
    float partial = 0.f;
#pragma unroll
    for (int v = 0; v < 8; ++v) {
        const int t = rbase + half * 8 + v;   // tap index in D-matrix layout
        if (t < TAPS) {
            const int i  = t / KW, j = t - i * KW;
            const int yy = py + i - 3, xx = px + j - 3;
            const float xv = (yy >= 0 && yy < HW && xx >= 0 && xx < HW)
                                 ? xpl[yy * HW + xx] : 0.f;
            partial += (acc[v] + b2c[t]) * xv;
        }
    }
    return partial;
}

// ---------------- Kernel 4: fused conv2 (WMMA f32) + involution --------------
// Block: 256 thr = 8 waves. blockIdx = (tileGroup, c, b). Wave w -> pixel tile.
__global__ void __launch_bounds__(256)
involution_fused(const float* __restrict__ x,
                 const float* __restrict__ w2,
                 const float* __restrict__ b2,
                 const float* __restrict__ hpre,
                 const float* __restrict__ scl,   // sclsft[0..63]
                 const float* __restrict__ sft,   // sclsft[64..127]
                 float* __restrict__ out) {
    __shared__ float w2c[64 * 65];        // 64 tap-rows x 64 k, stride 65 (bank-pad)
    __shared__ float b2c[64];
    __shared__ float xpl[NPIX];           // one 56x56 channel plane
    __shared__ float hT[8][CK * 16];      // per-wave 64x16 B tiles (BN+ReLU applied)

    const int tid = threadIdx.x;
    const int tg  = blockIdx.x;           // 0..24
    const int c   = blockIdx.y;           // 0..255
    const int b   = blockIdx.z;           // 0..3

    // ---- stage w2 rows for this channel (rows 49..63 zeroed) ----
    for (int e = tid; e < 64 * 64; e += 256) {
        const int r = e >> 6, k = e & 63;
        w2c[r * 65 + k] = (r < TAPS) ? w2[(size_t)(c * TAPS + r) * CK + k] : 0.f;
    }
    for (int e = tid; e < 64; e += 256)
        b2c[e] = (e < TAPS) ? b2[c * TAPS + e] : 0.f;
    // ---- stage x channel plane ----
    for (int e = tid; e < NPIX; e += 256)
        xpl[e] = x[(size_t)(b * CCH + c) * NPIX + e];
    // ---- stage h tiles (apply BN scale/shift + ReLU) ----
    const int pbase = tg * 128;           // 8 tiles * 16 pixels
#pragma unroll
    for (int it = 0; it < 32; ++it) {
        const int e = it * 256 + tid;     // < 8192
        const int k = e >> 7;             // 0..63
        const int q = e & 127;            // 0..127
        const int p = pbase + q;
        float v = 0.f;
        if (p < NPIX) {
            v = hpre[(size_t)(b * CK + k) * NPIX + p];
            v = v * scl[k] + sft[k];
            v = v > 0.f ? v : 0.f;
        }
        hT[q >> 4][k * 16 + (q & 15)] = v;
    }
    __syncthreads();

    // ---- per-wave WMMA GEMM: wgt[64 taps x 16 px] = w2c[64x64] * hT[64x16] ----
    const int wave = tid >> 5;
    const int lane = tid & 31;
    const int tile = tg * 8 + wave;
    if (tile < NTILE) {                   // wave-uniform -> EXEC stays all-1s
        const int half = lane >> 4;       // 0|1
        const int colN = lane & 15;       // pixel column / A row
        const float* hW = &hT[wave][0];

        v8f acc0 = {0,0,0,0,0,0,0,0};
        v8f acc1 = {0,0,0,0,0,0,0,0};
        v8f acc2 = {0,0,0,0,0,0,0,0};
        v8f acc3 = {0,0,0,0,0,0,0,0};

#pragma unroll
        for (int ks = 0; ks < 16; ++ks) { // K = 64 in steps of 4
            const int k0 = ks * 4;
            const int kk = k0 + 2 * half; // this lane's K pair
            // B fragment (4x16): vgpr0 = row kk, vgpr1 = row kk+1, col = colN
            v2f bf;
            bf.x = hW[kk * 16 + colN];
            bf.y = hW[(kk + 1) * 16 + colN];
            // A fragments (16x4): row M = rbase+colN, cols kk, kk+1
            v2f a0, a1, a2, a3;
            a0.x = w2c[( 0 + colN) * 65 + kk]; a0.y = w2c[( 0 + colN) * 65 + kk + 1];
            a1.x = w2c[(16 + colN) * 65 + kk]; a1.y = w2c[(16 + colN) * 65 + kk + 1];
            a2.x = w2c[(32 + colN) * 65 + kk]; a2.y = w2c[(32 + colN) * 65 + kk + 1];
            a3.x = w2c[(48 + colN) * 65 + kk]; a3.y = w2c[(48 + colN) * 65 + kk + 1];
            acc0 = __builtin_amdgcn_wmma_f32_16x16x4_f32(false, a0, false, bf, (short)0, acc0, false, false);
            acc1 = __builtin_amdgcn_wmma_f32_16x16x4_f32(false, a1, false, bf, (short)0, acc1, false, false);
            acc2 = __builtin_amdgcn_wmma_f32_16x16x4_f32(false, a2, false, bf, (short)0, acc2, false, false);
            acc3 = __builtin_amdgcn_wmma_f32_16x16x4_f32(false, a3, false, bf, (short)0, acc3, false, false);
        }

        // ---- involution: contract 49 taps against shifted x window ----
        const int p  = tile * 16 + colN;
        const int py = p / HW;
        const int px = p - py * HW;
        float partial = 0.f;
        partial += tap_accum(acc0,  0, half, py, px, xpl, b2c);
        partial += tap_accum(acc1, 16, half, py, px, xpl, b2c);
        partial += tap_accum(acc2, 32, half, py, px, xpl, b2c);
        partial += tap_accum(acc3, 48, half, py, px, xpl, b2c);
        // combine the two lane halves (taps split across lanes l and l+16)
        partial += __shfl_xor(partial, 16, 32);
        if (half == 0)
            out[(size_t)(b * CCH + c) * NPIX + p] = partial;
    }
}

// ------------------------------- launcher ------------------------------------
extern "C" void kernel_launch(void* const* d_in, const int* in_sizes, int n_in,
                              void* d_out, int out_size, void* d_ws, size_t ws_size,
                              hipStream_t stream) {
    const float* x     = (const float*)d_in[0];
    const float* w1    = (const float*)d_in[1];
    const float* b1    = (const float*)d_in[2];
    const float* gamma = (const float*)d_in[3];
    const float* beta  = (const float*)d_in[4];
    const float* w2    = (const float*)d_in[5];
    const float* b2    = (const float*)d_in[6];
    float*       out   = (float*)d_out;

    float* ws     = (float*)d_ws;
    float* stats  = ws;            // 128 floats
    float* sclsft = ws + 128;      // 128 floats
    float* hpre   = ws + 256;      // 4*64*3136 = 802816 floats (~3.2 MB)

    involution_conv1<<<dim3(3136), dim3(256), 0, stream>>>(x, w1, b1, hpre);
    involution_bnreduce<<<dim3(64), dim3(256), 0, stream>>>(hpre, stats);
    involution_bnfinalize<<<dim3(1), dim3(64), 0, stream>>>(stats, gamma, beta, sclsft);
    involution_fused<<<dim3(25, CCH, BATCH), dim3(256), 0, stream>>>(
        x, w2, b2, hpre, sclsft, sclsft + CK, out);
}